// TripletLossMultipleMargins_25890062860767
// MI455X (gfx1250) — compile-verified
//
#include <hip/hip_runtime.h>

typedef float v2f __attribute__((ext_vector_type(2)));
typedef float v8f __attribute__((ext_vector_type(8)));

#define TLD 64          // feature dim (reference: D = 64)
#define TILE_ROWS 16    // rows per wave tile (WMMA M)

// Kernel 1: each wave processes 16-row tiles.
//   E[m][k] = (a-p)^2 - (a-n)^2 elementwise
//   C init  = margin per row (broadcast across N)
//   16 x V_WMMA_F32_16X16X4_F32 with B = ones  ->  c[r] = dp - dn + margin
// then relu + reduce, block partial to workspace.
// All streaming loads are non-temporal (768 MB touched once; don't thrash L2).
__global__ __launch_bounds__(256) void triplet_wmma_kernel(
    const float* __restrict__ A, const float* __restrict__ P,
    const float* __restrict__ N, const int* __restrict__ AL,
    const int* __restrict__ NL, const float* __restrict__ MM,
    float* __restrict__ partial, int numTiles, int nClasses)
{
    const int lane        = threadIdx.x & 31;
    const int waveInBlock = threadIdx.x >> 5;
    const int wavesPerBlk = blockDim.x >> 5;
    const int globalWave  = blockIdx.x * wavesPerBlk + waveInBlock;
    const int totalWaves  = gridDim.x * wavesPerBlk;

    const int rowInTile = lane & 15;   // M index this lane feeds (A layout)
    const int half      = lane >> 4;   // 0 -> K={0,1}, 1 -> K={2,3}
    const int colSub    = half * 2;    // column sub-offset inside each K-chunk

    v2f ones; ones.x = 1.0f; ones.y = 1.0f;

    float wsum = 0.0f;

    for (int tile = globalWave; tile < numTiles; tile += totalWaves) {
        const int rowBase = tile * TILE_ROWS;
        const int row     = rowBase + rowInTile;

        const float* a = A + row * TLD + colSub;
        const float* p = P + row * TLD + colSub;
        const float* n = N + row * TLD + colSub;

        // lane-parallel margin gather: lane L fetches margin for row rowBase+L%16
        int   al  = __builtin_nontemporal_load(AL + rowBase + rowInTile);
        int   nl  = __builtin_nontemporal_load(NL + rowBase + rowInTile);
        float mrg = MM[al * nClasses + nl];   // 400 B table: keep cached (RT)

        // C accumulator pre-loaded with margins:
        // VGPR r, lanes 0-15 -> row r ; lanes 16-31 -> row r+8 (replicated over N)
        v8f c;
#pragma unroll
        for (int r = 0; r < 8; ++r) {
            c[r] = __shfl(mrg, r + half * 8, 32);
        }

        // 16 K-chunks of 4: c += E(16x4) * ones(4x16)
#pragma unroll
        for (int j = 0; j < TLD / 4; ++j) {
            v2f av = __builtin_nontemporal_load((const v2f*)(a + j * 4));
            v2f pv = __builtin_nontemporal_load((const v2f*)(p + j * 4));
            v2f nv = __builtin_nontemporal_load((const v2f*)(n + j * 4));
            v2f dp, dn, e;
            dp.x = av.x - pv.x;  dp.y = av.y - pv.y;
            dn.x = av.x - nv.x;  dn.y = av.y - nv.y;
            e.x  = dp.x * dp.x - dn.x * dn.x;
            e.y  = dp.y * dp.y - dn.y * dn.y;
            c = __builtin_amdgcn_wmma_f32_16x16x4_f32(
                    /*neg_a=*/false, e, /*neg_b=*/false, ones,
                    /*c_mod=*/(short)0, c, /*reuse_a=*/false, /*reuse_b=*/false);
        }

        // relu + per-lane sum of its half's 8 rows (identical across lanes of a half)
        float s = 0.0f;
#pragma unroll
        for (int r = 0; r < 8; ++r) s += fmaxf(c[r], 0.0f);

        // tile total = rows 0-7 (lane 0) + rows 8-15 (lane 16)
        wsum += __shfl(s, 0, 32) + __shfl(s, 16, 32);
    }

    // block reduction of wave partials (deterministic)
    __shared__ float lds[32];
    if (lane == 0) lds[waveInBlock] = wsum;
    __syncthreads();
    if (threadIdx.x == 0) {
        float b = 0.0f;
        for (int w = 0; w < wavesPerBlk; ++w) b += lds[w];
        partial[blockIdx.x] = b;
    }
}

// Kernel 2: reduce block partials, scale by 1/B.
__global__ __launch_bounds__(256) void triplet_reduce_kernel(
    const float* __restrict__ partial, float* __restrict__ out,
    int nPartials, float invB)
{
    __shared__ float lds[256];
    float s = 0.0f;
    for (int i = threadIdx.x; i < nPartials; i += blockDim.x) s += partial[i];
    lds[threadIdx.x] = s;
    __syncthreads();
    for (int stride = 128; stride > 0; stride >>= 1) {
        if ((int)threadIdx.x < stride) lds[threadIdx.x] += lds[threadIdx.x + stride];
        __syncthreads();
    }
    if (threadIdx.x == 0) out[0] = lds[0] * invB;
}

extern "C" void kernel_launch(void* const* d_in, const int* in_sizes, int n_in,
                              void* d_out, int out_size, void* d_ws, size_t ws_size,
                              hipStream_t stream) {
    const float* anchor   = (const float*)d_in[0];
    const float* positive = (const float*)d_in[1];
    const float* negative = (const float*)d_in[2];
    const int*   alabel   = (const int*)d_in[3];
    const int*   nlabel   = (const int*)d_in[4];
    const float* marginM  = (const float*)d_in[5];

    const int B = in_sizes[3];                 // 1048576
    int nClasses = 1;
    while (nClasses * nClasses < in_sizes[5]) ++nClasses;  // 10

    const int numTiles = B / TILE_ROWS;        // B % 16 == 0 for this problem
    const int blocks   = 2048;                 // 8 waves/block -> 16384 waves

    float* partial = (float*)d_ws;             // 2048 floats of scratch

    triplet_wmma_kernel<<<blocks, 256, 0, stream>>>(
        anchor, positive, negative, alabel, nlabel, marginM,
        partial, numTiles, nClasses);

    triplet_reduce_kernel<<<1, 256, 0, stream>>>(
        partial, (float*)d_out, blocks, 1.0f / (float)B);
}